// Group_61065845014588
// MI455X (gfx1250) — compile-verified
//
#include <hip/hip_runtime.h>

// Point-cloud KNN grouping for MI455X (gfx1250, wave32).
//  - sqrdist tiles via V_WMMA_F32_16X16X4_F32 (rank-4 f32 GEMM)
//  - point staging via Tensor Data Mover (tensor_load_to_lds), double-buffered
//  - top-32 selection via wave32 bitonic merge networks (K == warpSize)

typedef __attribute__((ext_vector_type(2))) float v2f;
typedef __attribute__((ext_vector_type(8))) float v8f;
typedef __attribute__((ext_vector_type(4))) unsigned int u32x4;
typedef __attribute__((ext_vector_type(8))) int i32x8;
typedef __attribute__((ext_vector_type(4))) int i32x4;

#define NBATCH 8
#define NPTS   16384
#define NCH    6
#define NGROUP 512
#define GSIZE  32
#define CTILE  16    // centers per workgroup (one WMMA M-tile)
#define PTILE  512   // points staged per TDM chunk
#define NTHR   256   // 8 waves of 32
#define NWAVES 8

#if __has_builtin(__builtin_amdgcn_tensor_load_to_lds)
#define HAVE_TDM 1
#else
#define HAVE_TDM 0
#endif

__device__ __forceinline__ bool pair_lt(float ad, int ai, float bd, int bi) {
    // strict order on (dist, index) — matches jax.lax.top_k tie-breaking
    return (ad < bd) || ((ad == bd) && (ai < bi));
}

#if HAVE_TDM
// 1-D contiguous DMA: copies n_elems f32 from gptr to the LDS address in lds_addr.
// D# layout per CDNA5 ISA.
// This toolchain exposes the 6-arg builtin:
//   (u32x4 g0, i32x8 g1, i32x4 g2, i32x4 g3, i32x8 extra, i32 cpol)
__device__ __forceinline__ void tdm_load_1d(unsigned int lds_addr,
                                            const void* gptr,
                                            unsigned int n_elems,
                                            unsigned int tensor_elems) {
    unsigned long long ga = (unsigned long long)(uintptr_t)gptr;
    u32x4 g0;
    g0[0] = 1u;                                   // count=1, user descriptor
    g0[1] = lds_addr;                             // lds_addr (bytes)
    g0[2] = (unsigned int)ga;                     // global_addr[31:0]
    g0[3] = (unsigned int)((ga >> 32) & 0x01FFFFFFu) | 0x80000000u; // [56:32] | type=2
    i32x8 g1;
    g1[0] = (int)(2u << 16);                      // wg_mask=0, data_size=4B
    g1[1] = (int)((tensor_elems & 0xFFFFu) << 16);          // tensor_dim0 lo16
    g1[2] = (int)((tensor_elems >> 16) & 0xFFFFu) | (1 << 16); // dim0 hi16 | tensor_dim1=1
    g1[3] = (int)((n_elems & 0xFFFFu) << 16);     // tensor_dim1 hi=0 | tile_dim0
    g1[4] = 1;                                    // tile_dim1=1, tile_dim2=0
    g1[5] = (int)tensor_elems;                    // tensor_dim0_stride lo32
    g1[6] = 0;                                    // stride0 hi16 | stride1 lo16
    g1[7] = 0;                                    // stride1 hi32
    i32x4 z4 = {0, 0, 0, 0};                      // groups 2/3: unused (<=2D)
    i32x8 z8 = {0, 0, 0, 0, 0, 0, 0, 0};          // extra descriptor words: inert
    __builtin_amdgcn_tensor_load_to_lds(g0, g1, z4, z4, z8, 0);
}
#endif

__global__ __launch_bounds__(NTHR) void group_knn_kernel(
        const float* __restrict__ pts,        // [B, N, 6]
        const int*   __restrict__ fps_index,  // [B, 512]
        float*       __restrict__ out_nb,     // [B, 512, 32, 6]
        float*       __restrict__ out_feats)  // [B, 512, 6]
{
    __shared__ float s_raw[2][PTILE * NCH];  // raw 6-f32 point rows, 2x12KB
    __shared__ float s_ctr[CTILE][4];        // cx, cy, cz, ||c||^2
    __shared__ float s_dist[CTILE][PTILE];   // sqrdist tile (32KB)

    const int b    = blockIdx.y;
    const int cg   = blockIdx.x;             // center-tile index
    const int t    = threadIdx.x;
    const int wave = t >> 5;
    const int lane = t & 31;
    const int m    = lane & 15;
    const bool hi  = lane >= 16;

    const float* batch_pts = pts + (size_t)b * NPTS * NCH;

    // ---- load 16 centers; emit feats output directly ----
    if (t < CTILE) {
        int g  = cg * CTILE + t;
        int fi = fps_index[b * NGROUP + g];
        const float* p = batch_pts + (size_t)fi * NCH;
        float x = p[0], y = p[1], z = p[2];
        s_ctr[t][0] = x; s_ctr[t][1] = y; s_ctr[t][2] = z;
        s_ctr[t][3] = x * x + y * y + z * z;
        float* fo = out_feats + ((size_t)b * NGROUP + g) * NCH;
        fo[0] = x; fo[1] = y; fo[2] = z;
        fo[3] = p[3]; fo[4] = p[4]; fo[5] = p[5];
    }
    __syncthreads();

    // ---- A fragment: 16x4 f32, rows = (-2cx, -2cy, -2cz, ||c||^2) ----
    // Layout (ISA 7.12.2): lane 0-15 -> M=lane, first VGPR holds K0 (lanes 0-15) or K2 (lanes 16-31), second holds K1 or K3
    v2f afrag;
    afrag.x = hi ? (-2.0f * s_ctr[m][2]) : (-2.0f * s_ctr[m][0]);  // K=2 : K=0
    afrag.y = hi ? (         s_ctr[m][3]) : (-2.0f * s_ctr[m][1]); // K=3 : K=1

#if HAVE_TDM
    const unsigned int lds_raw[2] = {
        (unsigned int)(uintptr_t)(void*)&s_raw[0][0],
        (unsigned int)(uintptr_t)(void*)&s_raw[1][0]
    };
    // prologue: DMA chunk 0 into buffer 0
    if (wave == 0)
        tdm_load_1d(lds_raw[0], batch_pts, PTILE * NCH, NPTS * NCH);
#endif

    // Running top-32 per owned center, lane-sorted ascending (lane l = l-th NN)
    float bestd0 = __builtin_inff(), bestd1 = __builtin_inff();
    int   besti0 = 0x7fffffff,       besti1 = 0x7fffffff;

    int buf = 0;
    for (int bp = 0; bp < NPTS; bp += PTILE) {
#if HAVE_TDM
        if (wave == 0) __builtin_amdgcn_s_wait_tensorcnt(0); // chunk bp landed
        __syncthreads();                                     // publish to all waves
        if (wave == 0 && bp + PTILE < NPTS)                  // overlap DMA of next
            tdm_load_1d(lds_raw[buf ^ 1],
                        batch_pts + (size_t)(bp + PTILE) * NCH,
                        PTILE * NCH, (NPTS - bp - PTILE) * NCH);
#else
        for (int i = t; i < PTILE * NCH; i += NTHR)
            s_raw[buf][i] = batch_pts[(size_t)bp * NCH + i];
        __syncthreads();
#endif
        const float* rb = &s_raw[buf][0];

        // ---- WMMA: 16 centers x 16 points per op -> sqrdist tile in LDS ----
        for (int st = wave; st < PTILE / 16; st += NWAVES) {
            int n = st * 16 + m;
            float px = rb[n * NCH + 0];
            float py = rb[n * NCH + 1];
            float pz = rb[n * NCH + 2];
            float xn = px * px + py * py + pz * pz;
            // B fragment: 4x16, cols = (x, y, z, 1); symmetric layout to A
            v2f bfrag;
            bfrag.x = hi ? pz   : px;    // K=2 : K=0
            bfrag.y = hi ? 1.0f : py;    // K=3 : K=1
            v8f cfrag = {xn, xn, xn, xn, xn, xn, xn, xn};   // + ||x_n||^2
            v8f dtile = __builtin_amdgcn_wmma_f32_16x16x4_f32(
                false, afrag, false, bfrag, (short)0, cfrag, false, false);
            // D layout: VGPR r -> row r (lanes 0-15) / row r+8 (lanes 16-31)
            int mbase = hi ? 8 : 0;
            #pragma unroll
            for (int r = 0; r < 8; ++r)
                s_dist[mbase + r][n] = dtile[r];
        }
        __syncthreads();

        // ---- top-32 maintenance: each wave owns 2 centers ----
        #pragma unroll
        for (int cc = 0; cc < 2; ++cc) {
            int c   = wave * 2 + cc;
            float bd = cc ? bestd1 : bestd0;
            int   bi = cc ? besti1 : besti0;
            for (int ch = 0; ch < PTILE / 32; ++ch) {
                float d  = s_dist[c][ch * 32 + lane];
                int   id = bp + ch * 32 + lane;
                // wave-uniform prune: skip chunk if nothing can enter the list
                float curmax = __shfl(bd, 31, 32);
                if (!__any(d <= curmax)) continue;

                // bitonic sort of the 32 candidates, ascending across lanes
                #pragma unroll
                for (int k = 2; k <= 32; k <<= 1) {
                    #pragma unroll
                    for (int j = k >> 1; j >= 1; j >>= 1) {
                        float od = __shfl_xor(d, j, 32);
                        int   oi = __shfl_xor(id, j, 32);
                        bool dirUp = ((lane & k) == 0);
                        bool lower = ((lane & j) == 0);
                        bool oless = pair_lt(od, oi, d, id);
                        bool keep_other = (lower == dirUp) ? oless : !oless;
                        if (keep_other) { d = od; id = oi; }
                    }
                }
                // reversed elementwise merge against the running sorted list
                float rd = __shfl_xor(d, 31, 32);
                int   ri = __shfl_xor(id, 31, 32);
                if (pair_lt(rd, ri, bd, bi)) { bd = rd; bi = ri; }
                // clean the resulting bitonic sequence (5 steps)
                #pragma unroll
                for (int j = 16; j >= 1; j >>= 1) {
                    float od = __shfl_xor(bd, j, 32);
                    int   oi = __shfl_xor(bi, j, 32);
                    bool lower = ((lane & j) == 0);
                    bool oless = pair_lt(od, oi, bd, bi);
                    bool keep_other = lower ? oless : !oless;
                    if (keep_other) { bd = od; bi = oi; }
                }
            }
            if (cc) { bestd1 = bd; besti1 = bi; }
            else    { bestd0 = bd; besti0 = bi; }
        }
        __syncthreads();
        buf ^= 1;
    }

    // ---- emit neighborhood: lane l writes the l-th nearest neighbor ----
    #pragma unroll
    for (int cc = 0; cc < 2; ++cc) {
        int c  = wave * 2 + cc;
        int bi = cc ? besti1 : besti0;
        int g  = cg * CTILE + c;
        const float* p = batch_pts + (size_t)bi * NCH;
        float* no = out_nb + (((size_t)b * NGROUP + g) * GSIZE + lane) * NCH;
        no[0] = p[0] - s_ctr[c][0];
        no[1] = p[1] - s_ctr[c][1];
        no[2] = p[2] - s_ctr[c][2];
        no[3] = p[3]; no[4] = p[4]; no[5] = p[5];
    }
}

extern "C" void kernel_launch(void* const* d_in, const int* in_sizes, int n_in,
                              void* d_out, int out_size, void* d_ws, size_t ws_size,
                              hipStream_t stream) {
    (void)in_sizes; (void)n_in; (void)out_size; (void)d_ws; (void)ws_size;
    const float* pts = (const float*)d_in[0];
    const int*   fps = (const int*)d_in[1];
    float* out_nb    = (float*)d_out;
    float* out_feats = out_nb + (size_t)NBATCH * NGROUP * GSIZE * NCH;

    dim3 grid(NGROUP / CTILE, NBATCH);   // 32 x 8 = 256 workgroups
    group_knn_kernel<<<grid, NTHR, 0, stream>>>(pts, fps, out_nb, out_feats);
}